// pose3d_calibration_8461085573568
// MI455X (gfx1250) — compile-verified
//
#include <hip/hip_runtime.h>

typedef __attribute__((ext_vector_type(2))) float v2f;
typedef __attribute__((ext_vector_type(8))) float v8f;

#define BLOCK_THREADS 256
#define WAVES_PER_BLOCK (BLOCK_THREADS / 32)
#define GRID_BLOCKS 512

// One wave == one batch element per iteration.
// WMMA f32 16x16x4 computes P = R^T * (pose3d - C) for all 15 joints in one shot:
//   A (16x4)  = R^T   rows i (3 used), cols j (3 used)
//   B (4x16)  = X     rows j (3 used), cols k = joint (15 used)
//   D         : lane k (k<15, lo half) holds D[0]=Px[k], D[1]=Py[k], D[2]=Pz[k]
// All loads are unconditional with clamped per-lane offsets (no EXEC juggling);
// garbage lands only in unused rows/columns of the WMMA operands.
__global__ __launch_bounds__(BLOCK_THREADS)
void pose_main_kernel(const float* __restrict__ p3d,
                      const float* __restrict__ p2d,
                      const float* __restrict__ Rd,
                      const float* __restrict__ Cd,
                      float* __restrict__ partial,
                      int nB)
{
    const int lane = threadIdx.x & 31;
    const int wv   = threadIdx.x >> 5;
    const int gw   = blockIdx.x * WAVES_PER_BLOCK + wv;
    const int nw   = gridDim.x * WAVES_PER_BLOCK;
    const int m    = lane & 15;
    const bool hi  = lane >= 16;

    const int mc3  = (m < 2)  ? m : 2;    // clamp for R gathers
    const int mc15 = (m < 14) ? m : 14;   // clamp for pose gathers

    // Loop-invariant per-lane element offsets (relative to each batch record).
    const int offA0 = (hi ? 6 : 0) + mc3;    // R^T col j=0 / j=2
    const int offA1 = 3 + mc3;               // R^T col j=1 (lo lanes only)
    const int offB0 = (hi ? 30 : 0) + mc15;  // pose3d row 0 / row 2
    const int offB1 = 15 + mc15;             // pose3d row 1 (lo lanes only)
    const int offC0 = hi ? 2 : 0;            // C component matching offB0
    const int offQu = mc15;                  // pose_2d u row
    const int offQv = 15 + mc15;             // pose_2d v row

    // Bone tables: lanes 0..5 -> left bones, lanes 6..11 -> right bones.
    const int lb0[6] = {1, 5, 6, 14, 11, 12};
    const int lb1[6] = {5, 6, 7, 11, 12, 13};
    const int rb0[6] = {1, 2, 3, 14,  8,  9};
    const int rb1[6] = {2, 3, 4,  8,  9, 10};
    int j0 = 0, j1 = 0;
    if (lane < 6)       { j0 = lb0[lane];     j1 = lb1[lane];     }
    else if (lane < 12) { j0 = rb0[lane - 6]; j1 = rb1[lane - 6]; }

    const bool projLane = (!hi) && (m < 15);
    const bool symLane  = (lane < 6);

    float accS = 0.0f;
    float accP = 0.0f;

    #pragma unroll 2
    for (int b = gw; b < nB; b += nw) {
        const float* P = p3d + (size_t)b * 45;
        const float* R = Rd  + (size_t)b * 9;
        const float* C = Cd  + (size_t)b * 3;
        const float* Q = p2d + (size_t)b * 30;

        // 8 unconditional gathers -> one load clause, one wait.
        float a0  = R[offA0];
        float a1l = R[offA1];
        float b0r = P[offB0];
        float b1r = P[offB1];
        float c0  = C[offC0];
        float c1  = C[1];
        float qu  = Q[offQu];
        float qv  = Q[offQv];

        // Only mandatory mask: A[0..2][3] must be 0 (kills garbage B row K=3).
        float a1 = hi ? 0.0f : a1l;
        float b0 = b0r - c0;
        float b1 = b1r - c1;          // hi-lane value unused by WMMA (A col 3 == 0)

        v2f A;  A[0]  = a0; A[1]  = a1;
        v2f Bm; Bm[0] = b0; Bm[1] = b1;
        v8f Cz = {};
        v8f D = __builtin_amdgcn_wmma_f32_16x16x4_f32(
            false, A, false, Bm, (short)0, Cz, false, false);

        // ---- projection error (branchless mask) ----
        float inv = 512.0f * __builtin_amdgcn_rcpf(D[2]);   // FLEN / z, ~1 ulp
        float u   = __builtin_fmaf(D[0], inv, 512.0f);
        float v   = __builtin_fmaf(D[1], inv, 256.0f);
        float eu  = projLane ? (u - qu) : 0.0f;
        float ev  = projLane ? (v - qv) : 0.0f;
        accP = __builtin_fmaf(eu, eu, accP);
        accP = __builtin_fmaf(ev, ev, accP);

        // ---- bone symmetry from already-loaded X (C cancels in differences) ----
        // X0[k] in b0 lane k, X1[k] in b1 lane k, X2[k] in b0 lane k+16.
        float x0 = __shfl(b0, j0,      32);
        float y0 = __shfl(b1, j0,      32);
        float z0 = __shfl(b0, j0 + 16, 32);
        float x1 = __shfl(b0, j1,      32);
        float y1 = __shfl(b1, j1,      32);
        float z1 = __shfl(b0, j1 + 16, 32);
        float dx = x0 - x1, dy = y0 - y1, dz = z0 - z1;
        float len  = __builtin_fmaf(dx, dx, __builtin_fmaf(dy, dy, dz * dz));
        float lenR = __shfl(len, lane + 6, 32);   // lanes 0..5 fetch right-bone len
        float d    = symLane ? (len - lenR) : 0.0f;
        accS = __builtin_fmaf(d, d, accS);
    }

    // ---- wave32 reduction ----
    #pragma unroll
    for (int off = 16; off > 0; off >>= 1) {
        accS += __shfl_down(accS, off, 32);
        accP += __shfl_down(accP, off, 32);
    }

    __shared__ float sS[WAVES_PER_BLOCK];
    __shared__ float sP[WAVES_PER_BLOCK];
    if (lane == 0) { sS[wv] = accS; sP[wv] = accP; }
    __syncthreads();
    if (threadIdx.x == 0) {
        float ts = 0.0f, tp = 0.0f;
        #pragma unroll
        for (int i = 0; i < WAVES_PER_BLOCK; ++i) { ts += sS[i]; tp += sP[i]; }
        partial[2 * blockIdx.x]     = ts;
        partial[2 * blockIdx.x + 1] = tp;
    }
}

__global__ __launch_bounds__(BLOCK_THREADS)
void pose_reduce_kernel(const float* __restrict__ partial, int nPart,
                        float* __restrict__ out, float invS, float invP)
{
    __shared__ float sS[BLOCK_THREADS];
    __shared__ float sP[BLOCK_THREADS];
    float s = 0.0f, p = 0.0f;
    for (int i = threadIdx.x; i < nPart; i += BLOCK_THREADS) {
        s += partial[2 * i];
        p += partial[2 * i + 1];
    }
    sS[threadIdx.x] = s;
    sP[threadIdx.x] = p;
    __syncthreads();
    for (int off = BLOCK_THREADS / 2; off > 0; off >>= 1) {
        if (threadIdx.x < off) {
            sS[threadIdx.x] += sS[threadIdx.x + off];
            sP[threadIdx.x] += sP[threadIdx.x + off];
        }
        __syncthreads();
    }
    if (threadIdx.x == 0) {
        out[0] = sS[0] * invS;   // sym
        out[1] = sP[0] * invP;   // proj
    }
}

extern "C" void kernel_launch(void* const* d_in, const int* in_sizes, int n_in,
                              void* d_out, int out_size, void* d_ws, size_t ws_size,
                              hipStream_t stream)
{
    const float* p3d = (const float*)d_in[0];   // (B,3,15)
    const float* p2d = (const float*)d_in[1];   // (B,2,15)
    const float* Rd  = (const float*)d_in[2];   // (B,3,3)
    const float* Cd  = (const float*)d_in[3];   // (B,3,1)
    float* out      = (float*)d_out;            // 2 floats: [sym, proj]
    float* partial  = (float*)d_ws;             // GRID_BLOCKS * 2 floats

    const int nB = in_sizes[0] / 45;

    pose_main_kernel<<<GRID_BLOCKS, BLOCK_THREADS, 0, stream>>>(
        p3d, p2d, Rd, Cd, partial, nB);

    const float invS = 1.0f / (6.0f  * (float)nB);
    const float invP = 1.0f / (30.0f * (float)nB);
    pose_reduce_kernel<<<1, BLOCK_THREADS, 0, stream>>>(
        partial, GRID_BLOCKS, out, invS, invP);
}